// GCNModel_55585466745361
// MI455X (gfx1250) — compile-verified
//
#include <hip/hip_runtime.h>
#include <math.h>

#define DIM 128

typedef float v2f __attribute__((ext_vector_type(2)));
typedef float v8f __attribute__((ext_vector_type(8)));

// ---------------------------------------------------------------- degree
__global__ void deg_scatter_kernel(const int* __restrict__ col,
                                   float* __restrict__ deg, int nE) {
    int e = blockIdx.x * blockDim.x + threadIdx.x;
    if (e < nE) atomicAdd(&deg[col[e]], 1.0f);
}

__global__ void rsqrt_kernel(float* __restrict__ deg, int nN) {
    int n = blockIdx.x * blockDim.x + threadIdx.x;
    if (n < nN) deg[n] = rsqrtf(deg[n] + 1.0f);
}

// ---------------------------------------------------------------- GEMM: H = X @ W  (fp32 WMMA 16x16x4)
// block = 256 threads = 8 waves; each block: 16 rows x 128 cols; wave w -> cols [16w,16w+16)
__global__ __launch_bounds__(256) void gemm_wmma_kernel(const float* __restrict__ X,
                                                        const float* __restrict__ W,
                                                        float* __restrict__ H, int nRows) {
    __shared__ float xs[16 * 132];  // pad 132: bank stride 4 -> conflict-free A reads
    const int tid  = threadIdx.x;
    const int wave = tid >> 5;
    const int lane = tid & 31;
    const int row0 = blockIdx.x * 16;

    // stage 16x128 X tile (512 float4)
    for (int i = tid; i < 512; i += 256) {
        int r = i >> 5, c4 = i & 31;
        int gr = row0 + r;
        float4 v = make_float4(0.f, 0.f, 0.f, 0.f);
        if (gr < nRows) v = ((const float4*)(X + (size_t)gr * DIM))[c4];
        float* dst = &xs[r * 132 + c4 * 4];
        dst[0] = v.x; dst[1] = v.y; dst[2] = v.z; dst[3] = v.w;
    }
    __syncthreads();

    const int m     = lane & 15;          // M (for A) and N (for B/C) index
    const int khalf = (lane >> 4) << 1;   // lanes 0-15: K+0/1, lanes 16-31: K+2/3
    const int ncol  = wave * 16 + m;      // this lane's output column

    v8f acc = {};
#pragma unroll
    for (int kb = 0; kb < DIM; kb += 4) {
        v2f a, b;
        a.x = xs[m * 132 + kb + khalf];
        a.y = xs[m * 132 + kb + khalf + 1];
        b.x = W[(size_t)(kb + khalf)     * DIM + ncol];
        b.y = W[(size_t)(kb + khalf + 1) * DIM + ncol];
        acc = __builtin_amdgcn_wmma_f32_16x16x4_f32(
            /*neg_a=*/false, a, /*neg_b=*/false, b,
            /*c_mod=*/(short)0, acc, /*reuse_a=*/false, /*reuse_b=*/false);
    }

    // C layout: VGPR j -> M = j + 8*(lane>=16), N = lane&15
    const int mbase = (lane >> 4) << 3;
#pragma unroll
    for (int j = 0; j < 8; ++j) {
        int gr = row0 + mbase + j;
        if (gr < nRows) H[(size_t)gr * DIM + ncol] = acc[j];
    }
}

// ---------------------------------------------------------------- edge scatter: agg[col] += h[row]*norm
// 32 threads per edge, float4 per thread
__global__ void edge_scatter_kernel(const float* __restrict__ H,
                                    const int* __restrict__ row,
                                    const int* __restrict__ col,
                                    const float* __restrict__ dis,
                                    float* __restrict__ agg, int nE) {
    unsigned idx = blockIdx.x * blockDim.x + threadIdx.x;
    unsigned e = idx >> 5;
    int c = (int)(idx & 31u);
    if (e >= (unsigned)nE) return;
    int r = row[e], t = col[e];
    float norm = dis[r] * dis[t];
    float4 hv = ((const float4*)(H + (size_t)r * DIM))[c];
    float* dst = agg + (size_t)t * DIM + c * 4;
    atomicAdd(dst + 0, hv.x * norm);
    atomicAdd(dst + 1, hv.y * norm);
    atomicAdd(dst + 2, hv.z * norm);
    atomicAdd(dst + 3, hv.w * norm);
}

// ---------------------------------------------------------------- x_next = relu(agg + h*dis^2 + b)  (in-place on agg)
__global__ void finalize_kernel(const float* __restrict__ H,
                                const float* __restrict__ dis,
                                const float* __restrict__ bias,
                                float* __restrict__ agg, int nN) {
    int idx = blockIdx.x * blockDim.x + threadIdx.x;
    int n = idx >> 5, c = idx & 31;
    if (n >= nN) return;
    float d = dis[n];
    float d2 = d * d;
    float4 a  = ((const float4*)(agg + (size_t)n * DIM))[c];
    float4 h  = ((const float4*)(H   + (size_t)n * DIM))[c];
    float4 bb = ((const float4*)bias)[c];
    float4 o;
    o.x = fmaxf(fmaf(h.x, d2, a.x) + bb.x, 0.f);
    o.y = fmaxf(fmaf(h.y, d2, a.y) + bb.y, 0.f);
    o.z = fmaxf(fmaf(h.z, d2, a.z) + bb.z, 0.f);
    o.w = fmaxf(fmaf(h.w, d2, a.w) + bb.w, 0.f);
    ((float4*)(agg + (size_t)n * DIM))[c] = o;
}

// ---------------------------------------------------------------- graph pooling scatter
__global__ void pool_scatter_kernel(const float* __restrict__ H,
                                    const int* __restrict__ batch,
                                    float* __restrict__ pooled,
                                    float* __restrict__ counts, int nN) {
    int idx = blockIdx.x * blockDim.x + threadIdx.x;
    int n = idx >> 5, c = idx & 31;
    if (n >= nN) return;
    int g = batch[n];
    float4 hv = ((const float4*)(H + (size_t)n * DIM))[c];
    float* dst = pooled + (size_t)g * DIM + c * 4;
    atomicAdd(dst + 0, hv.x);
    atomicAdd(dst + 1, hv.y);
    atomicAdd(dst + 2, hv.z);
    atomicAdd(dst + 3, hv.w);
    if (c == 0) atomicAdd(&counts[g], 1.0f);
}

// ---------------------------------------------------------------- head: sigmoid((pooled/cnt) @ Wf + bf)
__global__ void head_kernel(const float* __restrict__ pooled,
                            const float* __restrict__ counts,
                            const float* __restrict__ Wf,
                            const float* __restrict__ bf,
                            float* __restrict__ out, int nG) {
    int g = blockIdx.x * blockDim.x + threadIdx.x;
    if (g >= nG) return;
    float inv = 1.0f / fmaxf(counts[g], 1.0f);
    float s = 0.0f;
    for (int d = 0; d < DIM; ++d) s += pooled[(size_t)g * DIM + d] * Wf[d];
    s = s * inv + bf[0];
    out[g] = 1.0f / (1.0f + __expf(-s));
}

// ----------------------------------------------------------------
extern "C" void kernel_launch(void* const* d_in, const int* in_sizes, int n_in,
                              void* d_out, int out_size, void* d_ws, size_t ws_size,
                              hipStream_t stream) {
    const float* x     = (const float*)d_in[0];
    const int*   ei    = (const int*)d_in[1];   // [2, E]: row then col
    const int*   batch = (const int*)d_in[2];
    const float* W1 = (const float*)d_in[3];
    const float* b1 = (const float*)d_in[4];
    const float* W2 = (const float*)d_in[5];
    const float* b2 = (const float*)d_in[6];
    const float* W3 = (const float*)d_in[7];
    const float* b3 = (const float*)d_in[8];
    const float* Wf = (const float*)d_in[9];
    const float* bf = (const float*)d_in[10];

    const int nN = in_sizes[0] / DIM;
    const int nE = in_sizes[1] / 2;
    const int nG = out_size;
    const int* rowIdx = ei;
    const int* colIdx = ei + nE;

    // workspace layout (floats, 16B-aligned chunks)
    float* ws     = (float*)d_ws;
    size_t nfeat  = (size_t)nN * DIM;
    float* deg    = ws;                         // nN (becomes deg_inv_sqrt in place)
    float* bufA   = ws + 50048;                 // h after GEMM
    float* bufB   = bufA + nfeat;               // agg / x buffers
    float* bufC   = bufB + nfeat;
    float* pooled = bufC + nfeat;               // nG*DIM
    float* counts = pooled + (size_t)nG * DIM;  // nG
    (void)ws_size; (void)n_in;

    // deg_inv_sqrt
    hipMemsetAsync(deg, 0, (size_t)nN * sizeof(float), stream);
    deg_scatter_kernel<<<(nE + 255) / 256, 256, 0, stream>>>(colIdx, deg, nE);
    rsqrt_kernel<<<(nN + 255) / 256, 256, 0, stream>>>(deg, nN);

    const unsigned gemmBlocks  = (unsigned)((nN + 15) / 16);
    const unsigned edgeBlocks  = (unsigned)(((size_t)nE * 32 + 255) / 256);
    const unsigned nodeBlocks  = (unsigned)(((size_t)nN * 32 + 255) / 256);

    const float* xin = x;
    float* aggBufs[3] = {bufB, bufC, bufB};
    const float* Ws[3] = {W1, W2, W3};
    const float* bs[3] = {b1, b2, b3};

    for (int l = 0; l < 3; ++l) {
        float* agg = aggBufs[l];
        gemm_wmma_kernel<<<gemmBlocks, 256, 0, stream>>>(xin, Ws[l], bufA, nN);
        hipMemsetAsync(agg, 0, nfeat * sizeof(float), stream);
        edge_scatter_kernel<<<edgeBlocks, 256, 0, stream>>>(bufA, rowIdx, colIdx, deg, agg, nE);
        finalize_kernel<<<nodeBlocks, 256, 0, stream>>>(bufA, deg, bs[l], agg, nN);
        xin = agg;
    }

    hipMemsetAsync(pooled, 0, ((size_t)nG * DIM + nG) * sizeof(float), stream);
    pool_scatter_kernel<<<nodeBlocks, 256, 0, stream>>>(xin, batch, pooled, counts, nN);
    head_kernel<<<(nG + 63) / 64, 64, 0, stream>>>(pooled, counts, Wf, bf, (float*)d_out, nG);
}